// PairwiseL1Loss_45664092291537
// MI455X (gfx1250) — compile-verified
//
#include <hip/hip_runtime.h>
#include <hip/hip_bf16.h>

// Pairwise mean-L1 loss:
//   Mx[i,j] = mean_k |x[i,k]-x[j,k]| ; My likewise
//   out[i]  = -sum_j Mx[i,j]*My[i,j]
//
// Phase 1 (VALU-bound): tiled LDS kernel computes P = Mx .* My  -> d_ws (256KB)
// Phase 2 (WMMA):       row-sum of P via V_WMMA_F32_16X16X4_F32 with B = ones

#define BATCH 256
#define DIM   4096
#define CHUNK 64
#define PITCH 68   // 64 + 4 pad: (PITCH mod 64)=4 -> conflict-free b128 LDS reads

typedef __attribute__((ext_vector_type(2))) float v2f;
typedef __attribute__((ext_vector_type(8))) float v8f;

// ---------------------------------------------------------------------------
// Kernel 1: grid (8,16); block 256 threads. Block computes a 32(i) x 16(j)
// tile of P. Each thread owns 2 pairs: (i0+ti, j0+tj) and (i0+ti+16, j0+tj).
// ---------------------------------------------------------------------------
__global__ __launch_bounds__(256) void pair_l1_prod_kernel(
    const float* __restrict__ x, const float* __restrict__ y,
    float* __restrict__ P)
{
  __shared__ float xi[32 * PITCH];  // 32 i-rows of x (current chunk)
  __shared__ float xj[16 * PITCH];  // 16 j-rows of x
  __shared__ float yi[32 * PITCH];
  __shared__ float yj[16 * PITCH];

  const int i0  = blockIdx.x * 32;
  const int j0  = blockIdx.y * 16;
  const int tid = threadIdx.x;
  const int tj  = tid & 15;   // j offset: lanes vary tj -> coalesced P stores
  const int ti  = tid >> 4;   // i offset (and ti+16)

  float sx0 = 0.f, sx1 = 0.f, sy0 = 0.f, sy1 = 0.f;

  for (int c0 = 0; c0 < DIM; c0 += CHUNK) {
    __syncthreads();
    // ---- stage chunk into LDS (float4 per access, fully coalesced) ----
    {
      // i-rows: 32 rows * 16 float4 = 512 float4 -> 2 per thread
      #pragma unroll
      for (int t = 0; t < 2; ++t) {
        int f   = tid + 256 * t;       // 0..511
        int row = f >> 4;              // 0..31
        int c4  = (f & 15) << 2;       // col (floats)
        float4 vx = *(const float4*)(x + (size_t)(i0 + row) * DIM + c0 + c4);
        float4 vy = *(const float4*)(y + (size_t)(i0 + row) * DIM + c0 + c4);
        *(float4*)(xi + row * PITCH + c4) = vx;
        *(float4*)(yi + row * PITCH + c4) = vy;
      }
      // j-rows: 16 rows * 16 float4 = 256 float4 -> 1 per thread
      int row = tid >> 4;              // 0..15
      int c4  = (tid & 15) << 2;
      float4 vx = *(const float4*)(x + (size_t)(j0 + row) * DIM + c0 + c4);
      float4 vy = *(const float4*)(y + (size_t)(j0 + row) * DIM + c0 + c4);
      *(float4*)(xj + row * PITCH + c4) = vx;
      *(float4*)(yj + row * PITCH + c4) = vy;
    }
    __syncthreads();

    const float* pxa = xi + ti * PITCH;
    const float* pxb = xi + (ti + 16) * PITCH;
    const float* pxj = xj + tj * PITCH;
    const float* pya = yi + ti * PITCH;
    const float* pyb = yi + (ti + 16) * PITCH;
    const float* pyj = yj + tj * PITCH;

    #pragma unroll
    for (int k = 0; k < CHUNK; k += 4) {
      float4 a0 = *(const float4*)(pxa + k);
      float4 a1 = *(const float4*)(pxb + k);
      float4 b  = *(const float4*)(pxj + k);
      // fabs folds into VOP3 |src| modifiers on the adds
      sx0 += (__builtin_fabsf(a0.x - b.x) + __builtin_fabsf(a0.y - b.y)) +
             (__builtin_fabsf(a0.z - b.z) + __builtin_fabsf(a0.w - b.w));
      sx1 += (__builtin_fabsf(a1.x - b.x) + __builtin_fabsf(a1.y - b.y)) +
             (__builtin_fabsf(a1.z - b.z) + __builtin_fabsf(a1.w - b.w));
      float4 c0v = *(const float4*)(pya + k);
      float4 c1v = *(const float4*)(pyb + k);
      float4 dv  = *(const float4*)(pyj + k);
      sy0 += (__builtin_fabsf(c0v.x - dv.x) + __builtin_fabsf(c0v.y - dv.y)) +
             (__builtin_fabsf(c0v.z - dv.z) + __builtin_fabsf(c0v.w - dv.w));
      sy1 += (__builtin_fabsf(c1v.x - dv.x) + __builtin_fabsf(c1v.y - dv.y)) +
             (__builtin_fabsf(c1v.z - dv.z) + __builtin_fabsf(c1v.w - dv.w));
    }
  }

  // mean scaling: 1/4096 exact in binary
  const float scale = (1.0f / (float)DIM) * (1.0f / (float)DIM);
  P[(size_t)(i0 + ti)      * BATCH + (j0 + tj)] = sx0 * sy0 * scale;
  P[(size_t)(i0 + ti + 16) * BATCH + (j0 + tj)] = sx1 * sy1 * scale;
}

// ---------------------------------------------------------------------------
// Kernel 2: out[i] = -sum_j P[i,j] via WMMA: D = A(16x4) * ones(4x16) + C,
// K-loop over 256 columns. 16 blocks x 1 wave (32 threads).
// A layout (32-bit 16x4): lane L -> M = L&15; VGPR0 holds K = (L<16?0:2),
// VGPR1 holds K+1.  C/D layout: VGPR r -> (lanes 0-15: M=r, N=lane),
// (lanes 16-31: M=r+8, N=lane-16).
// ---------------------------------------------------------------------------
__global__ __launch_bounds__(32) void rowsum_wmma_kernel(
    const float* __restrict__ P, float* __restrict__ out)
{
  const int i0   = blockIdx.x * 16;
  const int lane = threadIdx.x;        // 0..31
  const int M    = lane & 15;
  const int koff = (lane >> 4) * 2;    // 0 or 2

#if defined(__AMDGCN__) && __has_builtin(__builtin_amdgcn_wmma_f32_16x16x4_f32)
  v8f c = {};
  v2f b;
  b.x = 1.0f;
  b.y = 1.0f;
  const float* rowp = P + (size_t)(i0 + M) * BATCH;
  #pragma unroll 4
  for (int k = 0; k < BATCH; k += 4) {
    v2f a = *(const v2f*)(rowp + k + koff);   // 8B aligned (k+koff even)
    c = __builtin_amdgcn_wmma_f32_16x16x4_f32(
        /*neg_a=*/false, a, /*neg_b=*/false, b,
        /*c_mod=*/(short)0, c, /*reuse_a=*/false, /*reuse_b=*/false);
  }
  // every column of D holds the row sum; extract from N==0 lanes
  if (lane == 0) {
    #pragma unroll
    for (int v = 0; v < 8; ++v) out[i0 + v] = -c[v];
  } else if (lane == 16) {
    #pragma unroll
    for (int v = 0; v < 8; ++v) out[i0 + 8 + v] = -c[v];
  }
#else
  // host-pass / fallback path: plain row reduction
  if (lane < 16) {
    const float* rowp = P + (size_t)(i0 + lane) * BATCH;
    float s = 0.f;
    for (int k = 0; k < BATCH; ++k) s += rowp[k];
    out[i0 + lane] = -s;
  }
  (void)M; (void)koff;
#endif
}

// ---------------------------------------------------------------------------
extern "C" void kernel_launch(void* const* d_in, const int* in_sizes, int n_in,
                              void* d_out, int out_size, void* d_ws, size_t ws_size,
                              hipStream_t stream) {
  const float* x = (const float*)d_in[0];
  const float* y = (const float*)d_in[1];
  float* out = (float*)d_out;
  float* P   = (float*)d_ws;   // needs 256*256*4 = 256 KB scratch

  pair_l1_prod_kernel<<<dim3(8, 16), 256, 0, stream>>>(x, y, P);
  rowsum_wmma_kernel<<<16, 32, 0, stream>>>(P, out);
}